// DifferentiableBundleAdjustment_58231166599326
// MI455X (gfx1250) — compile-verified
//
#include <hip/hip_runtime.h>

typedef float v2f __attribute__((ext_vector_type(2)));
typedef float v8f __attribute__((ext_vector_type(8)));

#define S_LEN   128
#define OUT_DIM 15
#define IN_DIM  32

// ---------------------------------------------------------------------------
// Kernel 1: positions (out channels 0..2) via WMMA cumsum, one wave per batch.
//
// pos[s] = init_pos + sum_{t<s} 0.1*dba[t,0:3]
// Define in[r] = (r==0) ? 0 : 0.1*dba[r-1,0:3]; then pos = init + L_incl @ in,
// tiled 16 rows at a time:
//   d_tile  = L_incl @ in_tile + carry      (output tile)
//   carry  += Ones   @ in_tile              (inter-tile prefix carry)
// Each 16x16 @ 16x16 product = 4 chained V_WMMA_F32_16X16X4_F32 (K=16 -> 4x4).
// ---------------------------------------------------------------------------
__global__ __launch_bounds__(256)
void pos_wmma_kernel(const float* __restrict__ dba,
                     const float* __restrict__ gt,
                     float* __restrict__ out)
{
    const int lane  = threadIdx.x & 31;
    const int wave  = threadIdx.x >> 5;
    const int batch = blockIdx.x * 8 + wave;

    const int m    = lane & 15;   // A row (M) / B column (N) / D column (N)
    const int half = lane >> 4;   // lane group 0..15 vs 16..31

    // A matrices (constant, built from lane id).
    // f32 16x4 A layout: v0 holds K = 2*half, v1 holds K = 2*half+1 (ISA 7.12.2).
    v2f aL[4];
#pragma unroll
    for (int kk = 0; kk < 4; ++kk) {
        const int k0 = kk * 4 + 2 * half;
        aL[kk].x = (k0     <= m) ? 1.0f : 0.0f;   // inclusive lower-triangular
        aL[kk].y = (k0 + 1 <= m) ? 1.0f : 0.0f;
    }
    const v2f aOne = {1.0f, 1.0f};                 // all-ones (carry matrix)

    // init_pos broadcast into the 3 used columns of the accumulator output
    float ip = 0.0f;
    if (m < 3) ip = gt[batch * (S_LEN * OUT_DIM) + m];

    const float* drow = dba + (size_t)batch * S_LEN * IN_DIM;
    float*       orow = out + (size_t)batch * S_LEN * OUT_DIM;

    v8f carry = {0.0f, 0.0f, 0.0f, 0.0f, 0.0f, 0.0f, 0.0f, 0.0f};

#pragma unroll 1
    for (int tile = 0; tile < 8; ++tile) {
        v8f d = carry;
#pragma unroll
        for (int kk = 0; kk < 4; ++kk) {
            // B slice 4x16: row k = step within tile, col n = channel (0..2 used).
            // v0 holds row 2*half, v1 holds row 2*half+1 of this K-slice.
            const int r0  = tile * 16 + kk * 4 + 2 * half;   // global row, v0
            const int rr0 = (r0 > 0) ? (r0 - 1) : 0;         // shifted input idx
            const int rr1 = r0;                              // (r0+1)-1, always >=0
            // Unconditional, always-in-bounds loads (row has 32 floats) so EXEC
            // stays all-ones for the WMMAs; zero out unused cols afterwards.
            const float b0 = drow[rr0 * IN_DIM + m] * 0.1f;
            const float b1 = drow[rr1 * IN_DIM + m] * 0.1f;
            v2f bm;
            bm.x = (m < 3 && r0 > 0) ? b0 : 0.0f;
            bm.y = (m < 3)           ? b1 : 0.0f;

            d     = __builtin_amdgcn_wmma_f32_16x16x4_f32(false, aL[kk], false, bm,
                                                          (short)0, d,     false, false);
            carry = __builtin_amdgcn_wmma_f32_16x16x4_f32(false, aOne,   false, bm,
                                                          (short)0, carry, false, false);
        }
        // D layout: VGPR r -> row (step) r + 8*half, lane -> column (channel) m.
        if (m < 3) {
#pragma unroll
            for (int r = 0; r < 8; ++r) {
                const int s = tile * 16 + r + 8 * half;
                orow[s * OUT_DIM + m] = ip + d[r];
            }
        }
    }
}

// ---------------------------------------------------------------------------
// Kernel 2: quaternion chain (out channels 3..6) + zero tail (7..14).
// 4 lanes per batch, one quaternion component per lane; the 4-wide norm
// reduction is two shfl_xor hops. Sequential over S (non-associative chain).
// ---------------------------------------------------------------------------
__global__ __launch_bounds__(256)
void quat_kernel(const float* __restrict__ dba,
                 const float* __restrict__ gt,
                 float* __restrict__ out)
{
    const int gid   = blockIdx.x * blockDim.x + threadIdx.x;
    const int batch = gid >> 2;
    const int c     = gid & 3;

    const float* drow = dba + (size_t)batch * S_LEN * IN_DIM;
    float*       orow = out + (size_t)batch * S_LEN * OUT_DIM;

    float q = gt[batch * (S_LEN * OUT_DIM) + 3 + c];

    // step 0: initial quaternion + zero tail (group of 4 lanes covers 3..14)
    orow[3 + c]  = q;
    orow[7 + c]  = 0.0f;
    orow[11 + c] = 0.0f;

    for (int s = 1; s < S_LEN; ++s) {
        q += drow[(s - 1) * IN_DIM + 3 + c] * 0.1f;
        float n2 = q * q;
        n2 += __shfl_xor(n2, 1, 32);
        n2 += __shfl_xor(n2, 2, 32);
        q *= rsqrtf(n2);
        float* o = orow + s * OUT_DIM;
        o[3 + c]  = q;
        o[7 + c]  = 0.0f;
        o[11 + c] = 0.0f;
    }
}

extern "C" void kernel_launch(void* const* d_in, const int* in_sizes, int n_in,
                              void* d_out, int out_size, void* d_ws, size_t ws_size,
                              hipStream_t stream) {
    (void)in_sizes; (void)n_in; (void)out_size; (void)d_ws; (void)ws_size;
    const float* dba = (const float*)d_in[0];
    // d_in[1] (imu_measurements) is unused by the reference computation.
    const float* gt  = (const float*)d_in[2];
    float* out = (float*)d_out;

    // B = 4096 batches: 8 waves (batches) per 256-thread block -> 512 blocks.
    pos_wmma_kernel<<<512, 256, 0, stream>>>(dba, gt, out);
    // 4 lanes per batch -> 16384 threads -> 64 blocks of 256.
    quat_kernel<<<64, 256, 0, stream>>>(dba, gt, out);
}